// SpectConvWithAttention_57483842290074
// MI455X (gfx1250) — compile-verified
//
#include <hip/hip_runtime.h>

// SpectConv with attention, fused for MI455X (gfx1250, wave32).
//
// Fusion: sum-branch + attention-branch share the aggregation
//   agg[n,k,c] = sum_{e: dst=n} edge_attr[e,k]*(1+att[e]) * x[src[e],c]
// followed by one GEMM  out = [x | agg] @ [W4 ; W0..3]  + bias,
// executed with V_WMMA_F32_16X16X4_F32 (native f32 matrix path).
//
// edge_index treated as int32 per harness convention ("integer -> const int*";
// JAX without x64 canonicalizes jnp.int64 -> int32).

typedef __attribute__((ext_vector_type(2))) float v2f;
typedef __attribute__((ext_vector_type(8))) float v8f;

__device__ __forceinline__ void atomicMaxF32(float* addr, float val) {
    int* ia = (int*)addr;
    int old = __float_as_int(*addr);
    while (__int_as_float(old) < val) {
        int assumed = old;
        old = atomicCAS(ia, assumed, __float_as_int(val));
        if (old == assumed) break;
    }
}

// ---------------- init workspace: mx=-inf, den=0, agg=0 -------------------
__global__ void init_ws_kernel(float* __restrict__ mx, float* __restrict__ den,
                               float* __restrict__ agg, int n_nodes, int agg_elems) {
    int i = blockIdx.x * blockDim.x + threadIdx.x;
    if (i < agg_elems) agg[i] = 0.0f;
    if (i < n_nodes) { mx[i] = -__builtin_inff(); den[i] = 0.0f; }
}

// ---------------- s[n] = x[n,:] . att_vector ------------------------------
__global__ void node_dot_kernel(const float* __restrict__ x,
                                const float* __restrict__ attv,
                                float* __restrict__ s, int n_nodes) {
    int n = blockIdx.x * blockDim.x + threadIdx.x;
    if (n >= n_nodes) return;
    const float4* xr = (const float4*)(x + (long)n * 64);
    const float4* av = (const float4*)attv;
    float acc = 0.0f;
#pragma unroll
    for (int i = 0; i < 16; ++i) {
        float4 a = xr[i], b = av[i];
        acc += a.x * b.x + a.y * b.y + a.z * b.z + a.w * b.w;
    }
    s[n] = acc;
}

// ---------------- raw logit per edge + segment max (atomic) ---------------
__global__ void edge_logit_kernel(const float* __restrict__ edge_attr,
                                  const int* __restrict__ ei,
                                  const float* __restrict__ s,
                                  float* __restrict__ raw, float* __restrict__ mx,
                                  int n_edges) {
    int e = blockIdx.x * blockDim.x + threadIdx.x;
    if (e >= n_edges) return;
    float4 ea = ((const float4*)edge_attr)[e];
    float m = 0.25f * (ea.x + ea.y + ea.z + ea.w);   // msgs.mean(axis=1) coeff
    int src = ei[e];
    int dst = ei[n_edges + e];
    float r = m * s[src];
    raw[e] = r;
    atomicMaxF32(mx + dst, r);
}

// ---------------- exp(logit - max) + segment sum (atomic), in-place -------
__global__ void edge_exp_kernel(const int* __restrict__ ei,
                                float* __restrict__ raw_ex,
                                const float* __restrict__ mx,
                                float* __restrict__ den, int n_edges) {
    int e = blockIdx.x * blockDim.x + threadIdx.x;
    if (e >= n_edges) return;
    int dst = ei[n_edges + e];
    float v = __expf(raw_ex[e] - mx[dst]);
    raw_ex[e] = v;
    atomicAdd(den + dst, v);
}

// ------- att score + fused-coefficient scatter: one wave32 per edge -------
// agg[dst, k, c] += edge_attr[e,k]*(1+att) * x[src, c]; lane owns c=lane, lane+32.
__global__ void edge_scatter_kernel(const float* __restrict__ x,
                                    const float* __restrict__ edge_attr,
                                    const int* __restrict__ ei,
                                    const float* __restrict__ ex,
                                    const float* __restrict__ den,
                                    float* __restrict__ att_out,
                                    float* __restrict__ agg, int n_edges) {
    int gid  = blockIdx.x * blockDim.x + threadIdx.x;
    int e    = gid >> 5;
    int lane = gid & 31;
    if (e >= n_edges) return;                 // wave-uniform
    int src = ei[e];
    int dst = ei[n_edges + e];
    float att = ex[e] / (den[dst] + 1e-16f);  // PyG segment-softmax EPS
    if (lane == 0) att_out[e] = att;
    float4 ea = ((const float4*)edge_attr)[e];
    float g = 1.0f + att;                     // fuses sum-branch + attention-branch
    float c0 = ea.x * g, c1 = ea.y * g, c2 = ea.z * g, c3 = ea.w * g;
    const float* xr = x + (long)src * 64;
    float x0 = xr[lane], x1 = xr[lane + 32];
    float* ar = agg + (long)dst * 256;
    atomicAdd(ar +   0 + lane,      c0 * x0);
    atomicAdd(ar +   0 + lane + 32, c0 * x1);
    atomicAdd(ar +  64 + lane,      c1 * x0);
    atomicAdd(ar +  64 + lane + 32, c1 * x1);
    atomicAdd(ar + 128 + lane,      c2 * x0);
    atomicAdd(ar + 128 + lane + 32, c2 * x1);
    atomicAdd(ar + 192 + lane,      c3 * x0);
    atomicAdd(ar + 192 + lane + 32, c3 * x1);
}

// ---------------- final GEMM via V_WMMA_F32_16X16X4_F32 -------------------
// out[16 rows x 64 cols] per wave: A = [x_row(64) | agg_row(256)], B = [W4 ; W0..3].
// A frag (16x4 f32, ISA 7.12.2): lanes 0-15 -> K={j,j+1}, lanes 16-31 -> K={j+2,j+3}.
// B frag mirrors A; C/D: VGPR r holds rows r (lanes 0-15) and r+8 (lanes 16-31).
__global__ void gemm_wmma_kernel(const float* __restrict__ x,
                                 const float* __restrict__ agg,
                                 const float* __restrict__ weight,  // [5,64,64]
                                 const float* __restrict__ bias,
                                 float* __restrict__ out, int n_nodes, int n_tiles) {
    int wave = threadIdx.x >> 5;
    int lane = threadIdx.x & 31;
    int tile = blockIdx.x * (blockDim.x >> 5) + wave;
    if (tile >= n_tiles) return;              // wave-uniform: EXEC all-1s for WMMA
    int n0   = tile * 16;
    int half = lane >> 4;                     // 0: K pair {0,1}; 1: K pair {2,3}
    int l    = lane & 15;

    int row = n0 + l;
    int rcl = row < n_nodes ? row : n_nodes - 1;   // clamp for ragged tail

    v8f acc[4];
    acc[0] = (v8f)0.0f; acc[1] = (v8f)0.0f; acc[2] = (v8f)0.0f; acc[3] = (v8f)0.0f;

    // Phase 1: j in [0,64) -> A from x, B rows from weight[4] (selfconn).
    const float* xrow  = x + (long)rcl * 64 + 2 * half;
    const float* wself = weight + 4 * 64 * 64;
    for (int j = 0; j < 64; j += 4) {
        v2f a = *(const v2f*)(xrow + j);
        const float* wcol = wself + (j + 2 * half) * 64 + l;
#pragma unroll
        for (int t = 0; t < 4; ++t) {
            v2f b; b.x = wcol[t * 16]; b.y = wcol[t * 16 + 64];
            acc[t] = __builtin_amdgcn_wmma_f32_16x16x4_f32(
                false, a, false, b, (short)0, acc[t], false, false);
        }
    }
    // Phase 2: j in [0,256) -> A from agg row, B rows from weight[0:4] flat.
    const float* arow = agg + (long)rcl * 256 + 2 * half;
    for (int j = 0; j < 256; j += 4) {
        v2f a = *(const v2f*)(arow + j);
        const float* wcol = weight + (j + 2 * half) * 64 + l;
#pragma unroll
        for (int t = 0; t < 4; ++t) {
            v2f b; b.x = wcol[t * 16]; b.y = wcol[t * 16 + 64];
            acc[t] = __builtin_amdgcn_wmma_f32_16x16x4_f32(
                false, a, false, b, (short)0, acc[t], false, false);
        }
    }
    // Epilogue: C/D layout -> rows n0 + r + 8*half, col = t*16 + l.
#pragma unroll
    for (int t = 0; t < 4; ++t) {
        int c = t * 16 + l;
        float bv = bias[c];
#pragma unroll
        for (int r = 0; r < 8; ++r) {
            int orow = n0 + r + 8 * half;
            if (orow < n_nodes) out[(long)orow * 64 + c] = acc[t][r] + bv;
        }
    }
}

extern "C" void kernel_launch(void* const* d_in, const int* in_sizes, int n_in,
                              void* d_out, int out_size, void* d_ws, size_t ws_size,
                              hipStream_t stream) {
    const float* x         = (const float*)d_in[0];  // [N,64]
    const float* edge_attr = (const float*)d_in[1];  // [E,4]
    const float* weight    = (const float*)d_in[2];  // [5,64,64]
    const float* bias      = (const float*)d_in[3];  // [64]
    const float* att_vec   = (const float*)d_in[4];  // [64,1]
    const int*   edge_idx  = (const int*)d_in[5];    // [2,E]

    const int n_nodes = in_sizes[0] / 64;
    const int n_edges = in_sizes[1] / 4;

    float* out_nodes = (float*)d_out;                            // [N,64]
    float* att_out   = (float*)d_out + (size_t)n_nodes * 64;     // [E]

    float* ws  = (float*)d_ws;
    float* s   = ws;  ws += n_nodes;      // x . att_vec per node
    float* mx  = ws;  ws += n_nodes;      // segment max
    float* den = ws;  ws += n_nodes;      // segment sum of exp
    float* exb = ws;  ws += n_edges;      // raw logits, then exp() in-place
    float* agg = ws;                      // [N, 4, 64] fused aggregation

    const int T = 256;
    int agg_elems = n_nodes * 256;

    init_ws_kernel<<<(agg_elems + T - 1) / T, T, 0, stream>>>(mx, den, agg, n_nodes, agg_elems);
    node_dot_kernel<<<(n_nodes + T - 1) / T, T, 0, stream>>>(x, att_vec, s, n_nodes);
    edge_logit_kernel<<<(n_edges + T - 1) / T, T, 0, stream>>>(edge_attr, edge_idx, s, exb, mx, n_edges);
    edge_exp_kernel<<<(n_edges + T - 1) / T, T, 0, stream>>>(edge_idx, exb, mx, den, n_edges);
    {
        long total = (long)n_edges * 32;   // one wave32 per edge
        edge_scatter_kernel<<<(int)((total + T - 1) / T), T, 0, stream>>>(
            x, edge_attr, edge_idx, exb, den, att_out, agg, n_edges);
    }
    {
        int n_tiles = (n_nodes + 15) / 16;
        int waves_per_block = 4;           // 128 threads
        int blocks = (n_tiles + waves_per_block - 1) / waves_per_block;
        gemm_wmma_kernel<<<blocks, waves_per_block * 32, 0, stream>>>(
            x, agg, weight, bias, out_nodes, n_nodes, n_tiles);
    }
}